// DeSCI_73237782331945
// MI455X (gfx1250) — compile-verified
//
#include <hip/hip_runtime.h>
#include <hip/hip_bf16.h>

// ---------------------------------------------------------------------------
// DeSCI / WNNM pipeline for MI455X (gfx1250, wave32).
// 3200 independent 64x60 WNNM problems, each LDS-resident; the reconstruction
// GEMM W = (A*diag(d)) * V^T uses V_WMMA_F32_16X16X4_F32.
// ---------------------------------------------------------------------------

#define IMG      256
#define PS       8          // patch size
#define PSTEP    2
#define PN       125        // (256-8)/2+1
#define LW       11         // local window
#define LSTEP    6
#define LN       20         // (125-11)/6+1
#define L2       400        // LN*LN
#define NF       8          // "f" dim (really kj of first unfold; harmless)
#define SIM      60
#define WNNM_C   1.41f

typedef float v2f __attribute__((ext_vector_type(2)));
typedef float v8f __attribute__((ext_vector_type(8)));

// spi buffer: [NF][64][PN][PN]  (8,000,000 floats = 32 MB)
#define SPI_ELEMS (NF * 64 * PN * PN)

// --- analytic fold counts ---------------------------------------------------
__device__ __forceinline__ int cnt_win(int i) {   // local-window fold (k=11,s=6, n=20)
    int c = 0;
    #pragma unroll
    for (int wi = 0; wi < LW; ++wi) {
        int r = i - wi;
        if (r >= 0 && (r % LSTEP) == 0 && (r / LSTEP) < LN) ++c;
    }
    return c;
}
__device__ __forceinline__ int cnt_img(int I) {   // image fold (k=8,s=2, n=125)
    int c = 0;
    #pragma unroll
    for (int ki = 0; ki < PS; ++ki) {
        int r = I - ki;
        if (r >= 0 && (r & 1) == 0 && (r >> 1) < PN) ++c;
    }
    return c;
}

// --- K0: build spi ----------------------------------------------------------
// spi[f][p][i][j] = input[c][0][2i+ki][2j+kj],  ch=p*8+f, c=ch>>6, ki=(ch>>3)&7, kj=ch&7
__global__ void k_spi(const float* __restrict__ in, float* __restrict__ spi) {
    int e = blockIdx.x * blockDim.x + threadIdx.x;
    if (e >= SPI_ELEMS) return;
    int j = e % PN;
    int i = (e / PN) % PN;
    int p = (e / (PN * PN)) % 64;
    int f = e / (64 * PN * PN);
    int ch = p * NF + f;
    int c  = ch >> 6;
    int ki = (ch >> 3) & 7;
    int kj = ch & 7;
    spi[e] = in[(c * IMG + (i * PSTEP + ki)) * IMG + (j * PSTEP + kj)];
}

// --- K1: distances + top-60 (largest, stable) -------------------------------
__global__ void k_topk(const float* __restrict__ spi, int* __restrict__ idxb) {
    int b = blockIdx.x;                // 3200 = NF*L2
    int f = b / L2, l = b % L2;
    int li = l / LN, lj = l % LN;
    int tid = threadIdx.x;             // 128 threads

    __shared__ float cen[64];
    __shared__ float dist[LW * LW];
    __shared__ float rv[128];
    __shared__ int   ri[128];

    int ci5 = li * LSTEP + 5, cj5 = lj * LSTEP + 5;  // center w=60 -> (5,5)
    if (tid < 64)
        cen[tid] = spi[((f * 64 + tid) * PN + ci5) * PN + cj5];
    __syncthreads();

    if (tid < LW * LW) {
        int wi = tid / LW, wj = tid % LW;
        int i5 = li * LSTEP + wi, j5 = lj * LSTEP + wj;
        const float* col = spi + ((size_t)f * 64 * PN + i5) * PN + j5;
        float d = 0.f;
        #pragma unroll 8
        for (int p = 0; p < 64; ++p) {
            float v = col[(size_t)p * PN * PN] - cen[p];
            d += v * v;
        }
        dist[tid] = d;                 // sqrt is monotone; compare squared
    }
    __syncthreads();

    int* out = idxb + (size_t)b * SIM;
    for (int kk = 0; kk < SIM; ++kk) {
        float v = (tid < LW * LW) ? dist[tid] : -1.f;
        rv[tid] = v; ri[tid] = tid;
        __syncthreads();
        for (int s = 64; s > 0; s >>= 1) {
            if (tid < s) {
                float vo = rv[tid + s]; int io = ri[tid + s];
                if (vo > rv[tid] || (vo == rv[tid] && io < ri[tid])) { rv[tid] = vo; ri[tid] = io; }
            }
            __syncthreads();
        }
        if (tid == 0) { out[kk] = ri[0]; dist[ri[0]] = -2.f; }
        __syncthreads();
    }
}

// --- K2: WNNM per group: gather -> center -> one-sided Jacobi SVD ->
//         shrink -> W = A*diag(d)*V^T (WMMA f32) -> atomic scatter of deltas --
__global__ void __launch_bounds__(256) k_wnnm(const float* __restrict__ spi,
                                              const int* __restrict__ idxb,
                                              const float* __restrict__ sigma,
                                              float* __restrict__ acc) {
    __shared__ float A[64][65];
    __shared__ float V[64][65];
    __shared__ float meanS[64];
    __shared__ float dS[64];
    __shared__ int   widx[64];
    __shared__ int   perm[64];

    int b = blockIdx.x;                // NF*L2
    int f = b / L2, l = b % L2;
    int li = l / LN, lj = l % LN;
    int tid  = threadIdx.x;
    int wave = tid >> 5;
    int lane = tid & 31;

    if (tid < SIM) widx[tid] = idxb[(size_t)b * SIM + tid];
    else if (tid < 64) widx[tid] = 0;
    if (tid < 64) perm[tid] = tid;
    __syncthreads();

    // gather X (64 rows p, 60 cols k; pad to 64 cols with 0); V = I
    for (int e = tid; e < 64 * 64; e += 256) {
        int p = e >> 6, k = e & 63;
        float v = 0.f;
        if (k < SIM) {
            int w = widx[k];
            int i5 = li * LSTEP + w / LW, j5 = lj * LSTEP + w % LW;
            v = spi[((f * 64 + p) * PN + i5) * PN + j5];
        }
        A[p][k] = v;
        V[p][k] = (p == k) ? 1.f : 0.f;
    }
    __syncthreads();

    if (tid < 64) {
        float m = 0.f;
        for (int k = 0; k < SIM; ++k) m += A[tid][k];
        meanS[tid] = m * (1.f / SIM);
    }
    __syncthreads();
    for (int e = tid; e < 64 * 64; e += 256) {
        int p = e >> 6, k = e & 63;
        if (k < SIM) A[p][k] -= meanS[p];
    }
    __syncthreads();

    // one-sided Jacobi: 8 sweeps x 63 round-robin rounds x 32 disjoint pairs
    for (int sweep = 0; sweep < 8; ++sweep) {
        for (int rnd = 0; rnd < 63; ++rnd) {
            for (int pj = wave; pj < 32; pj += 8) {
                int ci = perm[pj], cj = perm[63 - pj];
                float xi0 = A[lane][ci],      xj0 = A[lane][cj];
                float xi1 = A[lane + 32][ci], xj1 = A[lane + 32][cj];
                float aa = xi0 * xi0 + xi1 * xi1;
                float bb = xj0 * xj0 + xj1 * xj1;
                float cc = xi0 * xj0 + xi1 * xj1;
                #pragma unroll
                for (int off = 16; off > 0; off >>= 1) {
                    aa += __shfl_xor(aa, off);
                    bb += __shfl_xor(bb, off);
                    cc += __shfl_xor(cc, off);
                }
                if (fabsf(cc) > 1e-18f) {
                    float tau = (bb - aa) / (2.f * cc);
                    float t   = (tau >= 0.f ? 1.f : -1.f) /
                                (fabsf(tau) + sqrtf(1.f + tau * tau));
                    float cs  = 1.f / sqrtf(1.f + t * t);
                    float sn  = t * cs;
                    A[lane][ci]      = cs * xi0 - sn * xj0;
                    A[lane][cj]      = sn * xi0 + cs * xj0;
                    A[lane + 32][ci] = cs * xi1 - sn * xj1;
                    A[lane + 32][cj] = sn * xi1 + cs * xj1;
                    float vi0 = V[lane][ci],      vj0 = V[lane][cj];
                    float vi1 = V[lane + 32][ci], vj1 = V[lane + 32][cj];
                    V[lane][ci]      = cs * vi0 - sn * vj0;
                    V[lane][cj]      = sn * vi0 + cs * vj0;
                    V[lane + 32][ci] = cs * vi1 - sn * vj1;
                    V[lane + 32][cj] = sn * vi1 + cs * vj1;
                }
            }
            __syncthreads();
            if (tid == 0) {            // rotate round-robin permutation
                int tmp = perm[63];
                for (int i = 63; i >= 2; --i) perm[i] = perm[i - 1];
                perm[1] = tmp;
            }
            __syncthreads();
        }
    }

    // singular values + WNNM shrink -> d_j = Sx_j / S_j
    if (tid < 64) {
        float s2 = 0.f;
        for (int p = 0; p < 64; ++p) s2 += A[p][tid] * A[p][tid];
        float S  = sqrtf(s2);
        float ns = sigma[0];
        float C0 = WNNM_C * sqrtf((float)SIM) * 2.f * ns * ns;
        float e  = 1e-5f;
        float Sp = S + e;
        float Delta = Sp * Sp - 4.f * C0;
        float Sx = 0.f;
        if (Delta >= 0.f) {
            Sx = S - e + 0.5f * sqrtf(Delta);
            if (Sx < 0.f) Sx = 0.f;
        }
        dS[tid] = (S > 1e-12f) ? (Sx / S) : 0.f;
    }
    __syncthreads();
    for (int e = tid; e < 64 * 64; e += 256) {
        int p = e >> 6, k = e & 63;
        A[p][k] *= dS[k];              // A <- U*S*diag(d) rotated = X*V*diag(d)
    }
    __syncthreads();

    // W = A * V^T  (64x64x64) with V_WMMA_F32_16X16X4_F32; scatter W+mean deltas
    int N    = lane & 15;
    int half = lane >> 4;              // 0: lanes 0-15, 1: lanes 16-31
    for (int tile = wave; tile < 16; tile += 8) {
        int tm = tile >> 2, tn = tile & 3;
        v8f c = {};
        int mrow  = 16 * tm + (lane & 15);
        int bcol  = 16 * tn + (lane & 15);
        int qbase = half * 2;
        #pragma unroll
        for (int kk = 0; kk < 16; ++kk) {
            int j0 = 4 * kk + qbase;
            v2f a, bfr;
            a.x   = A[mrow][j0];
            a.y   = A[mrow][j0 + 1];
            bfr.x = V[bcol][j0];       // B[q][n] = V^T[q][n] = V[n][q]
            bfr.y = V[bcol][j0 + 1];
            c = __builtin_amdgcn_wmma_f32_16x16x4_f32(
                    false, a, false, bfr, (short)0, c, false, false);
        }
        int k = 16 * tn + N;
        if (k < SIM) {
            int w = widx[k];
            int i5 = li * LSTEP + w / LW, j5 = lj * LSTEP + w % LW;
            #pragma unroll
            for (int r = 0; r < 8; ++r) {
                int p = 16 * tm + r + 8 * half;
                float val = c[r] + meanS[p];   // delta = W[p][k] + mean[p]
                atomicAdd(&acc[((size_t)(f * 64 + p) * PN + i5) * PN + j5], val);
            }
        }
    }
}

// --- K3: patch_img = spi + acc / m2f (in place into spi) --------------------
__global__ void k_patch(float* __restrict__ spi, const float* __restrict__ acc) {
    int e = blockIdx.x * blockDim.x + threadIdx.x;
    if (e >= SPI_ELEMS) return;
    int j = e % PN;
    int i = (e / PN) % PN;
    int m = cnt_win(i) * cnt_win(j);
    if (m == 0) m = 1;
    spi[e] += acc[e] / (float)m;
}

// --- K4: final fold to image (analytic m1f) ---------------------------------
__global__ void k_fold(const float* __restrict__ pimg, float* __restrict__ out) {
    int t = blockIdx.x * blockDim.x + threadIdx.x;
    if (t >= NF * IMG * IMG) return;
    int J = t & 255;
    int I = (t >> 8) & 255;
    int c = t >> 16;
    float s = 0.f;
    #pragma unroll
    for (int ki = 0; ki < PS; ++ki) {
        int ri = I - ki;
        if (ri < 0 || (ri & 1) || (ri >> 1) >= PN) continue;
        int i = ri >> 1;
        #pragma unroll
        for (int kj = 0; kj < PS; ++kj) {
            int rj = J - kj;
            if (rj < 0 || (rj & 1) || (rj >> 1) >= PN) continue;
            int j = rj >> 1;
            int p = c * PS + ki;       // ch = p*8 + f, f = kj
            s += pimg[(((size_t)kj * 64 + p) * PN + i) * PN + j];
        }
    }
    int dm = cnt_img(I) * cnt_img(J);
    if (dm == 0) dm = 1;
    out[t] = s / (float)dm;
}

// ---------------------------------------------------------------------------
extern "C" void kernel_launch(void* const* d_in, const int* in_sizes, int n_in,
                              void* d_out, int out_size, void* d_ws, size_t ws_size,
                              hipStream_t stream) {
    (void)in_sizes; (void)n_in; (void)out_size; (void)ws_size;
    const float* input = (const float*)d_in[0];   // (8,1,256,256) f32
    const float* sigma = (const float*)d_in[1];   // (1,) f32
    float* out = (float*)d_out;                   // (8,1,256,256) f32

    char*  ws   = (char*)d_ws;
    float* spi  = (float*)ws;                         // 32,000,000 B
    float* acc  = (float*)(ws + 32000000);            // 32,000,000 B
    int*   idxb = (int*)(ws + 64000000);              //    768,000 B

    hipMemsetAsync(acc, 0, (size_t)SPI_ELEMS * sizeof(float), stream);

    k_spi  <<<(SPI_ELEMS + 255) / 256, 256, 0, stream>>>(input, spi);
    k_topk <<<NF * L2, 128, 0, stream>>>(spi, idxb);
    k_wnnm <<<NF * L2, 256, 0, stream>>>(spi, idxb, sigma, acc);
    k_patch<<<(SPI_ELEMS + 255) / 256, 256, 0, stream>>>(spi, acc);
    k_fold <<<(NF * IMG * IMG + 255) / 256, 256, 0, stream>>>(spi, out);
}